// SpatialConsistencyLoss_85280870629491
// MI455X (gfx1250) — compile-verified
//
#include <hip/hip_runtime.h>
#include <hip/hip_bf16.h>

typedef __attribute__((ext_vector_type(2))) float v2f;
typedef __attribute__((ext_vector_type(8))) float v8f;

#define NPTS 16384
#define DIM  256
#define KNN  8
#define NTILES (NPTS / 16)
#define NWAVES 4                       // waves per block in kernel 2
#define TPW (NTILES / NWAVES)          // column tiles per wave
#define SENT 0x7F800000FFFFFFFFull     // (+inf distance, idx -1): real keys always smaller

// ---------------------------------------------------------------------------
// Kernel 1: per-point prep.
//  - packed coords {x,y,z, x^2+y^2+z^2} for WMMA distance tiles
//  - feature inverse L2 norm (1 / max(||f||, 1e-12))
// One wave per point (256 f32 -> 8 elems/lane).
// ---------------------------------------------------------------------------
__global__ __launch_bounds__(256) void sc_prep_kernel(
    const float* __restrict__ feat,
    const float* __restrict__ coords,
    float4* __restrict__ px4,
    float* __restrict__ invn)
{
    int gtid = blockIdx.x * blockDim.x + threadIdx.x;
    int p    = gtid >> 5;
    int lane = threadIdx.x & 31;
    if (p >= NPTS) return;

    const float4* f4 = reinterpret_cast<const float4*>(feat) + (size_t)p * (DIM / 4);
    float4 a = f4[lane * 2 + 0];
    float4 b = f4[lane * 2 + 1];
    float s = a.x * a.x + a.y * a.y + a.z * a.z + a.w * a.w
            + b.x * b.x + b.y * b.y + b.z * b.z + b.w * b.w;
#pragma unroll
    for (int off = 16; off > 0; off >>= 1)
        s += __shfl_xor(s, off, 32);

    if (lane == 0) {
        invn[p] = 1.0f / fmaxf(sqrtf(s), 1e-12f);
        float x = coords[p * 3 + 0];
        float y = coords[p * 3 + 1];
        float z = coords[p * 3 + 2];
        px4[p] = make_float4(x, y, z, x * x + y * y + z * z);
    }
}

// Sorted (ascending) top-8 insertion, fully unrolled -> registers.
__device__ __forceinline__ void insert8(unsigned long long (&l)[8],
                                        unsigned long long key)
{
    if (key < l[7]) {
        l[7] = key;
#pragma unroll
        for (int s = 7; s > 0; --s) {
            unsigned long long a = l[s - 1], b = l[s];
            bool sw = b < a;
            l[s - 1] = sw ? b : a;
            l[s]     = sw ? a : b;
        }
    }
}

// ---------------------------------------------------------------------------
// Kernel 2: one block (4 waves / 128 threads) handles 16 rows.
//  Phase A: each wave scans NTILES/4 column tiles with V_WMMA_F32_16X16X4_F32;
//           per-lane top-8 of packed (d2_bits<<32 | col) keys, guarded by a
//           cheap f32 threshold so the common path is add+fma+cmp.
//  Phase B: two-stage LDS merge: 64 lane-lists/row -> 8 chunks -> top-8.
//  Phase C: cosine similarity, 4 rows per wave, deterministic combine.
// ---------------------------------------------------------------------------
__global__ __launch_bounds__(128) void sc_knn_cos_kernel(
    const float* __restrict__ feat,
    const float4* __restrict__ px4,
    const float* __restrict__ invn,
    float* __restrict__ partials)
{
    __shared__ unsigned long long stage1[16 * 64 * 8];   // [row][lane64][8]  64 KB
    __shared__ unsigned long long stage2[16 * 8 * 8];    // [row][chunk][8]    8 KB
    __shared__ int   knnS[16][KNN];
    __shared__ float wpart[NWAVES];

    const int tid  = threadIdx.x;
    const int wid  = tid >> 5;          // wave 0..3
    const int lane = tid & 31;
    const int cls  = lane & 15;         // column-in-tile / A-row-in-tile
    const int half = lane >> 4;
    const int rowbase  = blockIdx.x * 16;
    const int rbh      = rowbase + 8 * half;   // row of slot v is rbh + v
    const int selfTile = blockIdx.x;           // only tile that can contain self

    // ---- A operand: 16x4 f32 (K=3 coords padded with 0) ----
    float4 pr = px4[rowbase + cls];
    v2f A;
    A.x = half ? pr.z : pr.x;           // lanes 0-15: K=0,1 ; lanes 16-31: K=2,3
    A.y = half ? 0.0f : pr.y;

    float sqA[8];
#pragma unroll
    for (int v = 0; v < 8; ++v)
        sqA[v] = px4[rbh + v].w;

    unsigned long long list[8][8];
    float thrF[8];
#pragma unroll
    for (int v = 0; v < 8; ++v) {
        thrF[v] = __builtin_inff();
#pragma unroll
        for (int s = 0; s < 8; ++s)
            list[v][s] = SENT;
    }

    // ---- Phase A: software-pipelined WMMA scan over this wave's tiles ----
    float4 pc = px4[wid * 16 + cls];
#pragma unroll 1
    for (int i = 0; i < TPW; ++i) {
        const int t = wid + i * NWAVES;
        // prefetch next tile's column data (px4 has 64 entries of tail padding)
        float4 pcn = px4[(t + NWAVES) * 16 + cls];

        v2f B;
        B.x = half ? pc.z : pc.x;
        B.y = half ? 0.0f : pc.y;
        v8f C = {0.f, 0.f, 0.f, 0.f, 0.f, 0.f, 0.f, 0.f};
        C = __builtin_amdgcn_wmma_f32_16x16x4_f32(
                false, A, false, B, (short)0, C, false, false);

        const int n = (t << 4) + cls;   // this lane's column index
        const float base = pc.w;        // sq of column point
#pragma unroll
        for (int v = 0; v < 8; ++v) {
            float d2 = fmaf(-2.0f, C[v], sqA[v] + base);
            if (d2 <= thrF[v]) {                         // cheap f32 pre-filter
                unsigned long long key =
                    ((unsigned long long)__float_as_uint(fmaxf(d2, 0.0f)) << 32)
                    | (unsigned)n;
                if (t == selfTile && (n - rbh) == v)     // exclude self (uniform tile test)
                    key = SENT;
                if (key < list[v][7]) {                  // exact 64-bit tie-break
                    list[v][7] = key;
#pragma unroll
                    for (int s = 7; s > 0; --s) {
                        unsigned long long x = list[v][s - 1], y = list[v][s];
                        bool sw = y < x;
                        list[v][s - 1] = sw ? y : x;
                        list[v][s]     = sw ? x : y;
                    }
                    thrF[v] = __uint_as_float((unsigned)(list[v][7] >> 32));
                }
            }
        }
        pc = pcn;
    }

    // ---- Phase B: two-stage merge in LDS ----
    {
        const int l64 = wid * 16 + cls;                  // 0..63
#pragma unroll
        for (int v = 0; v < 8; ++v) {
            int r = v + 8 * half;
#pragma unroll
            for (int s = 0; s < 8; ++s)
                stage1[(r * 64 + l64) * 8 + s] = list[v][s];
        }
    }
    __syncthreads();

    {   // stage 2a: 128 threads, each merges 8 lane-lists (64 keys) of one row
        const int r = tid >> 3, c = tid & 7;
        unsigned long long best[8];
#pragma unroll
        for (int s = 0; s < 8; ++s) best[s] = SENT;
        for (int j = 0; j < 64; ++j)
            insert8(best, stage1[(r * 64 + c * 8 + (j >> 3)) * 8 + (j & 7)]);
#pragma unroll
        for (int s = 0; s < 8; ++s)
            stage2[(r * 8 + c) * 8 + s] = best[s];
    }
    __syncthreads();

    if (tid < 16) {   // stage 2b: one thread per row merges 64 keys -> top-8
        unsigned long long best[8];
#pragma unroll
        for (int s = 0; s < 8; ++s) best[s] = SENT;
        for (int j = 0; j < 64; ++j)
            insert8(best, stage2[tid * 64 + j]);
#pragma unroll
        for (int k = 0; k < KNN; ++k)
            knnS[tid][k] = (int)(unsigned)best[k];       // low 32 bits = index
    }
    __syncthreads();

    // ---- Phase C: cosine similarity (4 rows per wave) ----
    const float4* f4 = reinterpret_cast<const float4*>(feat);
    float acc = 0.0f;
#pragma unroll 1
    for (int rr = 0; rr < 4; ++rr) {
        int r   = wid * 4 + rr;
        int row = rowbase + r;
        float invi = invn[row];
        float4 m0 = f4[(size_t)row * 64 + lane * 2 + 0];
        float4 m1 = f4[(size_t)row * 64 + lane * 2 + 1];
#pragma unroll
        for (int k = 0; k < KNN; ++k) {
            int nb = knnS[r][k];
            float4 g0 = f4[(size_t)nb * 64 + lane * 2 + 0];
            float4 g1 = f4[(size_t)nb * 64 + lane * 2 + 1];
            float p = m0.x * g0.x + m0.y * g0.y + m0.z * g0.z + m0.w * g0.w
                    + m1.x * g1.x + m1.y * g1.y + m1.z * g1.z + m1.w * g1.w;
            acc = fmaf(p, invi * invn[nb], acc);
        }
    }
#pragma unroll
    for (int off = 16; off > 0; off >>= 1)
        acc += __shfl_xor(acc, off, 32);
    if (lane == 0) wpart[wid] = acc;
    __syncthreads();
    if (tid == 0)   // fixed-order combine -> deterministic
        partials[blockIdx.x] = wpart[0] + wpart[1] + wpart[2] + wpart[3];
}

// ---------------------------------------------------------------------------
// Kernel 3: deterministic fixed-order reduction of 1024 partials -> loss.
// ---------------------------------------------------------------------------
__global__ __launch_bounds__(256) void sc_finalize_kernel(
    const float* __restrict__ partials, float* __restrict__ out)
{
    __shared__ float s[256];
    int t = threadIdx.x;
    float v = partials[t] + partials[t + 256] + partials[t + 512] + partials[t + 768];
    s[t] = v;
    __syncthreads();
#pragma unroll
    for (int off = 128; off > 0; off >>= 1) {
        if (t < off) s[t] += s[t + off];
        __syncthreads();
    }
    if (t == 0)
        out[0] = 0.02f * (1.0f - s[0] / ((float)NPTS * (float)KNN));
}

extern "C" void kernel_launch(void* const* d_in, const int* in_sizes, int n_in,
                              void* d_out, int out_size, void* d_ws, size_t ws_size,
                              hipStream_t stream) {
    const float* feat   = (const float*)d_in[0];   // 16384 x 256 f32
    const float* coords = (const float*)d_in[1];   // 16384 x 3   f32
    float* out = (float*)d_out;

    char* ws = (char*)d_ws;
    // px4 gets 64 entries of tail padding for the software-pipeline prefetch
    float4* px4   = (float4*)(ws);
    float*  invn  = (float*)(ws + (size_t)(NPTS + 64) * sizeof(float4));
    float*  parts = (float*)(ws + (size_t)(NPTS + 64) * sizeof(float4)
                                + (size_t)NPTS * sizeof(float));

    // 1 wave per point: 16384 waves -> 2048 blocks of 256 threads
    sc_prep_kernel<<<NPTS / 8, 256, 0, stream>>>(feat, coords, px4, invn);

    // 16 rows per block, 4 waves split the column scan: 1024 blocks x 128
    sc_knn_cos_kernel<<<NTILES, 128, 0, stream>>>(feat, px4, invn, parts);

    // deterministic finalize
    sc_finalize_kernel<<<1, 256, 0, stream>>>(parts, out);
}